// SuperLoss_3461743640772
// MI455X (gfx1250) — compile-verified
//
#include <hip/hip_runtime.h>
#include <stdint.h>

// ---------------- problem constants ----------------
#define NROWS        8192
#define NCLASSES     32000
#define BLOCK        512
#define CHUNK        8000          // floats per TDM chunk
#define CHUNK_BYTES  (CHUNK * 4)   // 32000 B
#define NCHUNKS      4             // 4 * 8000 = 32000 floats = one row
#define TAU_F        10.373491181781864f   // ln(32000)
#define E_F          2.718281828459045f
#define LOG2E_F      1.4426950408889634f
#define LN2_F        0.6931471805599453f

typedef unsigned int v4u __attribute__((ext_vector_type(4)));
typedef int          v8i __attribute__((ext_vector_type(8)));
typedef int          v4i __attribute__((ext_vector_type(4)));

// Issue one TDM load: CHUNK contiguous fp32 from global -> LDS (byte offset lds_addr).
// D# group0: count=1 | lds_addr | global_addr[56:0] | type=2
// D# group1: data_size=2(4B), tensor_dim0=8000, tensor_dim1=1, tile_dim0=8000,
//            tile_dim1=1, tensor_dim0_stride=8000 (1D contiguous, fully in-bounds).
__device__ __forceinline__ void tdm_load_chunk(uint64_t gaddr, uint32_t lds_addr) {
    v4u g0;
    g0[0] = 1u;                                   // count=1 (valid descriptor)
    g0[1] = lds_addr;                             // lds_addr (bytes)
    g0[2] = (uint32_t)gaddr;                      // global_addr[31:0]
    g0[3] = ((uint32_t)(gaddr >> 32) & 0x01FFFFFFu) | (2u << 30); // [56:32] | type=2
    v8i g1;
    g1[0] = (int)(2u << 16);                      // data_size = 2 -> 4 bytes
    g1[1] = (int)((unsigned)CHUNK << 16);         // tensor_dim0[15:0] at bits 63:48
    g1[2] = (int)(1u << 16);                      // tensor_dim0 hi = 0, tensor_dim1 = 1
    g1[3] = (int)((unsigned)CHUNK << 16);         // tile_dim0 at bits 127:112
    g1[4] = 1;                                    // tile_dim1 = 1, tile_dim2 = 0
    g1[5] = (int)CHUNK;                           // tensor_dim0_stride[31:0]
    g1[6] = 0;                                    // stride hi / dim1_stride lo
    g1[7] = 0;
    v4i z4 = (v4i)0;
#if defined(__clang_major__) && (__clang_major__ >= 23)
    v8i z8 = (v8i)0;
    __builtin_amdgcn_tensor_load_to_lds(g0, g1, z4, z4, z8, 0);
#else
    __builtin_amdgcn_tensor_load_to_lds(g0, g1, z4, z4, 0);
#endif
}

// online-softmax merge of (m, s) pairs in base-2 domain
__device__ __forceinline__ void merge_ms(float& m, float& s, float om, float os) {
    float nm = fmaxf(m, om);
    s = s * exp2f(m - nm) + os * exp2f(om - nm);
    m = nm;
}

__global__ __launch_bounds__(BLOCK)
void superloss_rows(const float* __restrict__ logits,
                    const int*   __restrict__ targets,
                    float*       __restrict__ row_loss) {
    extern __shared__ char dynsmem[];      // 2 * CHUNK_BYTES double buffer
    __shared__ float red_m[BLOCK / 32];
    __shared__ float red_s[BLOCK / 32];

    const int row = blockIdx.x;
    const uint64_t gbase = (uint64_t)(uintptr_t)(const void*)(logits + (size_t)row * NCLASSES);
    const uint32_t lds0  = (uint32_t)(uintptr_t)(void*)&dynsmem[0]; // low 32 bits = LDS byte offset
    const uint32_t lds1  = lds0 + CHUNK_BYTES;

    if (threadIdx.x == 0) {                 // wave0 drives the TDM pipeline
        tdm_load_chunk(gbase, lds0);                       // chunk 0
        tdm_load_chunk(gbase + CHUNK_BYTES, lds1);         // chunk 1
    }

    float m = -__builtin_inff();
    float s = 0.0f;

    for (int c = 0; c < NCHUNKS; ++c) {
        if (threadIdx.x == 0) {
            // TDM ops of one wave complete in order: after issuing k chunks,
            // tensorcnt <= k-1-c guarantees chunk c has landed in LDS.
            if (c < NCHUNKS - 1) __builtin_amdgcn_s_wait_tensorcnt(1);
            else                 __builtin_amdgcn_s_wait_tensorcnt(0);
        }
        __syncthreads();   // chunk c visible to all waves

        const float4* p = (const float4*)(dynsmem + (size_t)(c & 1) * CHUNK_BYTES);
        for (int i = threadIdx.x; i < CHUNK / 4; i += BLOCK) {
            float4 v = p[i];                               // ds_load_b128
            float a0 = v.x * LOG2E_F, a1 = v.y * LOG2E_F;
            float a2 = v.z * LOG2E_F, a3 = v.w * LOG2E_F;
            float m4 = fmaxf(fmaxf(a0, a1), fmaxf(a2, a3));
            float nm = fmaxf(m, m4);
            s = s * exp2f(m - nm)
              + (exp2f(a0 - nm) + exp2f(a1 - nm))
              + (exp2f(a2 - nm) + exp2f(a3 - nm));
            m = nm;
        }
        __syncthreads();   // everyone done reading buffer (c&1)

        if (threadIdx.x == 0 && (c + 2) < NCHUNKS) {       // refill freed buffer
            tdm_load_chunk(gbase + (uint64_t)(c + 2) * CHUNK_BYTES,
                           (c & 1) ? lds1 : lds0);
        }
    }

    // intra-wave reduction (wave32)
#pragma unroll
    for (int off = 16; off >= 1; off >>= 1) {
        float om = __shfl_xor(m, off, 32);
        float os = __shfl_xor(s, off, 32);
        merge_ms(m, s, om, os);
    }
    const int lane = threadIdx.x & 31;
    const int wid  = threadIdx.x >> 5;
    if (lane == 0) { red_m[wid] = m; red_s[wid] = s; }
    __syncthreads();

    if (threadIdx.x == 0) {
        float M = red_m[0], S = red_s[0];
#pragma unroll
        for (int w = 1; w < BLOCK / 32; ++w) merge_ms(M, S, red_m[w], red_s[w]);

        // logZ = max + ln(sum e^{x-max}) ; (M,S) are in base-2 domain
        float logZ = LN2_F * (M + log2f(S));
        int   tgt  = targets[row];
        float xt   = logits[(size_t)row * NCLASSES + (size_t)tgt];
        float ce   = logZ - xt;

        // SuperLoss weighting: y = 0.5*max(-2/e, (ce - tau)); w = W0(y) via Halley
        float y = 0.5f * fmaxf(-2.0f / E_F, ce - TAU_F);
        float w;
        if (y > E_F) { float big = logf(y); w = big - logf(big); }
        else         { w = log1pf(fmaxf(y, -0.999f / E_F)); }
#pragma unroll 1
        for (int it = 0; it < 20; ++it) {
            float ew  = expf(w);
            float f   = w * ew - y;
            float wp1 = w + 1.0f;
            w = w - f / (ew * wp1 - (w + 2.0f) * f / (2.0f * wp1));
        }
        float sigma = expf(-w);
        row_loss[row] = (ce - TAU_F) * sigma + w * w;
    }
}

__global__ __launch_bounds__(256)
void superloss_reduce(const float* __restrict__ row_loss, float* __restrict__ out) {
    __shared__ float buf[8];
    float acc = 0.0f;
#pragma unroll 1
    for (int k = 0; k < NROWS / 256; ++k)          // fixed order -> deterministic
        acc += row_loss[threadIdx.x + k * 256];
#pragma unroll
    for (int off = 16; off >= 1; off >>= 1)
        acc += __shfl_xor(acc, off, 32);
    if ((threadIdx.x & 31) == 0) buf[threadIdx.x >> 5] = acc;
    __syncthreads();
    if (threadIdx.x == 0) {
        float t = 0.0f;
#pragma unroll
        for (int i = 0; i < 8; ++i) t += buf[i];
        *out = t / (float)NROWS;
    }
}

extern "C" void kernel_launch(void* const* d_in, const int* in_sizes, int n_in,
                              void* d_out, int out_size, void* d_ws, size_t ws_size,
                              hipStream_t stream) {
    (void)in_sizes; (void)n_in; (void)out_size; (void)ws_size;
    const float* logits  = (const float*)d_in[0];
    const int*   targets = (const int*)d_in[1];
    float*       out     = (float*)d_out;
    float*       ws      = (float*)d_ws;          // NROWS floats of scratch

    superloss_rows<<<NROWS, BLOCK, 2 * CHUNK_BYTES, stream>>>(logits, targets, ws);
    superloss_reduce<<<1, 256, 0, stream>>>(ws, out);
}